// CoPEGate_71210557768292
// MI455X (gfx1250) — compile-verified
//
#include <hip/hip_runtime.h>

// CoPE gate: out[b,h,t,s] = sigmoid(Q·K^T / 8) * (P·P^T / 8)
// B=2 H=12 T=2048 D=64. Write-BW bound (403 MB out). f32 WMMA 16x16x4.

typedef __attribute__((ext_vector_type(2))) float v2f;
typedef __attribute__((ext_vector_type(8))) float v8f;

#define T_DIM 2048
#define D_DIM 64
#define BH_DIM 24          // B*H
#define TILES_PER_ROW (T_DIM / 16)               // 128
#define TILES_PER_MAT (TILES_PER_ROW * TILES_PER_ROW) // 16384

__global__ __launch_bounds__(256) void cope_gate_kernel(
    const float* __restrict__ Q,     // [BH, T, D]
    const float* __restrict__ Km,    // [BH, T, D]
    const float* __restrict__ P,     // [T, D]
    float* __restrict__ Out)         // [BH, T, T]
{
    const int lane = threadIdx.x & 31;
    const int wave = threadIdx.x >> 5;

    // One wave per 16x16 output tile.
    const long wid   = (long)blockIdx.x * 8 + wave;
    const int  bh    = (int)(wid / TILES_PER_MAT);      // 0..23
    const int  tile  = (int)(wid % TILES_PER_MAT);
    const int  t0    = (tile / TILES_PER_ROW) * 16;     // query-row tile base
    const int  s0    = (tile % TILES_PER_ROW) * 16;     // key-row tile base
    if (bh >= BH_DIM) return;

    // A-operand layout (16x4 f32, ISA 7.12.2): lane L holds row M = L%16;
    // VGPR0 = K = 2*(L/16), VGPR1 = K = 2*(L/16)+1.  B mirrors with N = L%16.
    const int m  = lane & 15;
    const int kh = (lane >> 4) << 1;   // 0 or 2

    const float* qrow = Q  + ((long)bh * T_DIM + (t0 + m)) * D_DIM;
    const float* krow = Km + ((long)bh * T_DIM + (s0 + m)) * D_DIM;
    const float* prT  = P  + (long)(t0 + m) * D_DIM;
    const float* prS  = P  + (long)(s0 + m) * D_DIM;

    v8f acc_s = {};   // Q·K^T tile
    v8f acc_p = {};   // P·P^T tile

#pragma unroll
    for (int k = 0; k < D_DIM; k += 4) {
        v2f a  = *(const v2f*)(qrow + k + kh);
        v2f b  = *(const v2f*)(krow + k + kh);
        v2f pa = *(const v2f*)(prT  + k + kh);
        v2f pb = *(const v2f*)(prS  + k + kh);
        // (neg_a, A, neg_b, B, c_mod, C, reuse_a, reuse_b)
        acc_s = __builtin_amdgcn_wmma_f32_16x16x4_f32(
            false, a, false, b, (short)0, acc_s, false, false);
        acc_p = __builtin_amdgcn_wmma_f32_16x16x4_f32(
            false, pa, false, pb, (short)0, acc_p, false, false);
    }

    const float scale = 0.125f;        // 1/sqrt(64)
    // C/D layout: VGPR r -> row (r + 8*(lane/16)), col = lane%16.
    const int rowOff = (lane >> 4) * 8;
    float* outBase = Out + (long)bh * T_DIM * T_DIM;

#pragma unroll
    for (int r = 0; r < 8; ++r) {
        float sc   = acc_s[r] * scale;
        float gate = 1.0f / (1.0f + __expf(-sc));
        float bias = acc_p[r] * scale;
        const long row = t0 + rowOff + r;
        const long col = s0 + m;
        // Output (403 MB) is 2x the L2: bypass with non-temporal store so
        // Q/K/P stay L2-resident across their 128x reuse.
        __builtin_nontemporal_store(gate * bias, outBase + row * T_DIM + col);
    }
}

extern "C" void kernel_launch(void* const* d_in, const int* in_sizes, int n_in,
                              void* d_out, int out_size, void* d_ws, size_t ws_size,
                              hipStream_t stream) {
    const float* q = (const float*)d_in[0];   // (B,H,T,D) f32
    const float* k = (const float*)d_in[1];   // (B,H,T,D) f32
    const float* p = (const float*)d_in[2];   // (MAX_SEQ_LEN, D) f32, use first T rows
    float* out = (float*)d_out;               // (B,H,T,T) f32

    const long totalWaves = (long)BH_DIM * TILES_PER_MAT;   // 393216
    const int  blocks     = (int)(totalWaves / 8);          // 8 waves / 256-thread block
    cope_gate_kernel<<<blocks, 256, 0, stream>>>(q, k, p, out);
}